// GNNLayer_21938692948450
// MI455X (gfx1250) — compile-verified
//
#include <hip/hip_runtime.h>

typedef __attribute__((ext_vector_type(2))) float v2f;
typedef __attribute__((ext_vector_type(8))) float v8f;

#define D 128
#define LN_EPS 1e-5f

// ---------- Kernel 1: deg = 1 (self-loop contribution), Hagg = 0 ----------
__global__ void k_init(float* __restrict__ Hagg, float* __restrict__ deg, int N) {
    int i = blockIdx.x * blockDim.x + threadIdx.x;
    if (i < N * D) Hagg[i] = 0.0f;
    if (i < N)     deg[i]  = 1.0f;
}

// ---------- Kernel 2: degree over source (row) indices ----------
__global__ void k_degree(const long long* __restrict__ row,
                         float* __restrict__ deg, int E) {
    int e = blockIdx.x * blockDim.x + threadIdx.x;
    if (e < E) {
        int r = (int)row[e];
        __hip_atomic_fetch_add(&deg[r], 1.0f,
                               __ATOMIC_RELAXED, __HIP_MEMORY_SCOPE_AGENT);
    }
}

// ---------- Kernel 3: deg_inv = 1 / max(deg, 1) ----------
__global__ void k_recip(const float* __restrict__ deg,
                        float* __restrict__ deg_inv, int N) {
    int i = blockIdx.x * blockDim.x + threadIdx.x;
    if (i < N) deg_inv[i] = 1.0f / fmaxf(deg[i], 1.0f);
}

// ---------- Kernel 4: gather-scale-scatter, one wave per edge ----------
// lane covers 4 consecutive features (32 lanes * 4 = 128); float4 gather,
// 4x hardware fp32 atomic adds into L2-resident Hagg.
__global__ void k_scatter(const float* __restrict__ H,
                          const long long* __restrict__ row,
                          const long long* __restrict__ col,
                          const float* __restrict__ deg_inv,
                          float* __restrict__ Hagg,
                          int E, int N) {
    int wave = (int)((blockIdx.x * blockDim.x + threadIdx.x) >> 5);
    int lane = threadIdx.x & 31;
    int total = E + N;
    if (wave >= total) return;
    int r, c;
    if (wave < E) { r = (int)row[wave]; c = (int)col[wave]; }
    else          { r = c = wave - E; }                       // self-loop
    float s = deg_inv[r];
    float4 h = ((const float4*)(H + (size_t)r * D))[lane];
    float* o = Hagg + (size_t)c * D + lane * 4;
    __hip_atomic_fetch_add(o + 0, h.x * s, __ATOMIC_RELAXED, __HIP_MEMORY_SCOPE_AGENT);
    __hip_atomic_fetch_add(o + 1, h.y * s, __ATOMIC_RELAXED, __HIP_MEMORY_SCOPE_AGENT);
    __hip_atomic_fetch_add(o + 2, h.z * s, __ATOMIC_RELAXED, __HIP_MEMORY_SCOPE_AGENT);
    __hip_atomic_fetch_add(o + 3, h.w * s, __ATOMIC_RELAXED, __HIP_MEMORY_SCOPE_AGENT);
}

// ---------- Kernel 5: fused WMMA GEMM + bias + ReLU + LayerNorm ----------
// One 256-thread block (8 wave32) per 16-row strip. Wave w computes output
// cols [16w, 16w+16) with V_WMMA_F32_16X16X4_F32, K=128 in 32 steps.
__global__ void __launch_bounds__(256)
k_gemm_ln(const float* __restrict__ Hagg, const float* __restrict__ W,
          const float* __restrict__ bias, const float* __restrict__ gamma,
          const float* __restrict__ beta, float* __restrict__ out, int N) {
    __shared__ float lds[16 * D];   // A tile, later reused for the output tile

    const int t       = threadIdx.x;
    const int rowBase = blockIdx.x * 16;

    // --- stage 16x128 A tile into LDS (row-clamped for the tail block) ---
    {
        int idx = t * 8;                    // 256 threads * 8 floats = 2048
        int r   = idx >> 7;
        int c0  = idx & 127;
        int gr  = min(rowBase + r, N - 1);
        const float* src = Hagg + (size_t)gr * D + c0;
        float4 x0 = *(const float4*)(src);
        float4 x1 = *(const float4*)(src + 4);
        *(float4*)(&lds[idx])     = x0;
        *(float4*)(&lds[idx + 4]) = x1;
    }
    __syncthreads();

    const int wave    = t >> 5;
    const int lane    = t & 31;
    const int halfsel = lane >> 4;          // 0: K+{0,1}, 1: K+{2,3}
    const int mn      = lane & 15;          // A row / B col within tile
    const int tileN   = wave * 16;

    // B[k][o] = W[o][k]  (W is D_OUT x D_IN row-major) -> read W row tileN+mn
    const float* wrow = W + (size_t)(tileN + mn) * D;
    const float* arow = &lds[mn * D];

    v8f acc = {};
    #pragma unroll
    for (int kb = 0; kb < 32; ++kb) {
        int kk = kb * 4 + halfsel * 2;      // even -> 8B-aligned float2
        v2f a = { arow[kk], arow[kk + 1] };
        v2f b = { wrow[kk], wrow[kk + 1] };
        acc = __builtin_amdgcn_wmma_f32_16x16x4_f32(
                  /*neg_a=*/false, a, /*neg_b=*/false, b,
                  /*c_mod=*/(short)0, acc, /*reuse_a=*/false, /*reuse_b=*/false);
    }
    __syncthreads();

    // --- bias + ReLU, deposit 16x16 tile into LDS (C/D VGPR layout) ---
    float bv = bias[tileN + mn];
    #pragma unroll
    for (int v = 0; v < 8; ++v) {
        int m = halfsel ? (v + 8) : v;
        lds[m * D + tileN + mn] = fmaxf(acc[v] + bv, 0.0f);
    }
    __syncthreads();

    // --- LayerNorm: 16 lanes per row, 8 elements each, shfl_xor reduce ---
    const int r = t >> 4;
    const int g = t & 15;
    const float* rp = &lds[r * D + g * 8];
    float vals[8], s = 0.f, ss = 0.f;
    #pragma unroll
    for (int j = 0; j < 8; ++j) {
        float x = rp[j]; vals[j] = x; s += x; ss += x * x;
    }
    #pragma unroll
    for (int m = 1; m < 16; m <<= 1) {
        s  += __shfl_xor(s,  m, 16);
        ss += __shfl_xor(ss, m, 16);
    }
    float mu  = s * (1.0f / D);
    float var = ss * (1.0f / D) - mu * mu;
    float rs  = rsqrtf(var + LN_EPS);

    int grow = rowBase + r;
    if (grow < N) {
        float* op = out + (size_t)grow * D + g * 8;
        #pragma unroll
        for (int j = 0; j < 8; ++j) {
            int cidx = g * 8 + j;
            op[j] = (vals[j] - mu) * rs * gamma[cidx] + beta[cidx];
        }
    }
}

extern "C" void kernel_launch(void* const* d_in, const int* in_sizes, int n_in,
                              void* d_out, int out_size, void* d_ws, size_t ws_size,
                              hipStream_t stream) {
    const float*     H   = (const float*)d_in[0];
    const long long* ei  = (const long long*)d_in[1];   // (2, E) int64
    const float*     W   = (const float*)d_in[3];
    const float*     b   = (const float*)d_in[4];
    const float*     gmm = (const float*)d_in[5];
    const float*     bta = (const float*)d_in[6];
    float*           out = (float*)d_out;

    const int N = in_sizes[0] / D;
    const int E = in_sizes[1] / 2;
    const long long* row = ei;
    const long long* col = ei + E;

    // workspace: Hagg (N*128 f32) | deg (N f32) | deg_inv (N f32)  ~5.2 MB
    float* Hagg    = (float*)d_ws;
    float* deg     = Hagg + (size_t)N * D;
    float* deg_inv = deg + N;

    k_init   <<<(N * D + 255) / 256, 256, 0, stream>>>(Hagg, deg, N);
    k_degree <<<(E + 255) / 256,      256, 0, stream>>>(row, deg, E);
    k_recip  <<<(N + 255) / 256,      256, 0, stream>>>(deg, deg_inv, N);
    k_scatter<<<(E + N + 7) / 8,      256, 0, stream>>>(H, row, col, deg_inv,
                                                        Hagg, E, N);
    k_gemm_ln<<<(N + 15) / 16,        256, 0, stream>>>(Hagg, W, b, gmm, bta,
                                                        out, N);
}